// nUIV_rhs_68410239090964
// MI455X (gfx1250) — compile-verified
//
#include <hip/hip_runtime.h>
#include <hip/hip_bf16.h>
#include <stdint.h>

// Problem size (fixed by the reference).
#define NN 8192
#define THREADS 256
#define ROWS_PER_BLOCK 16
#define NBLOCKS (NN / ROWS_PER_BLOCK)   // 512
#define CHUNKS 8                        // 8 x float4 per thread per row = 8192 floats

typedef __attribute__((ext_vector_type(2))) float v2f;
typedef __attribute__((ext_vector_type(8))) float v8f;

// Issue 8 async b128 copies: global row -> this thread's private LDS region.
// GVS addressing: global address is the 64-bit base in saddr plus the 32-bit offset in vaddr; LDS address in vdst.
__device__ __forceinline__ void async_row_load(const float* __restrict__ ts, int row,
                                               uint32_t ldsBufByte, int tid) {
    const float* base = ts + (size_t)row * NN;   // uniform per block -> SGPR pair
    unsigned long long sbase = (unsigned long long)(uintptr_t)base;
#pragma unroll
    for (int j = 0; j < CHUNKS; ++j) {
        uint32_t colByte = (uint32_t)((j * 1024 + tid * 4) * 4);
        uint32_t lds = ldsBufByte + colByte;
        asm volatile("global_load_async_to_lds_b128 %0, %1, %2"
                     :
                     : "v"(lds), "v"(colByte), "s"(sbase)
                     : "memory");
    }
}

__global__ __launch_bounds__(THREADS) void fused_rows_kernel(
    const float* __restrict__ state, const float* __restrict__ ts,
    float* __restrict__ g_coupling, float* __restrict__ g_norm,
    float* __restrict__ g_diag) {
    __shared__ float s_buf[NN];        // 32 KB row staging (thread-private 32-float slices)
    __shared__ float s_red[THREADS];   // block reduction scratch
    __shared__ float s_misc[2];        // {diag^2, V[row]}

    const int tid  = threadIdx.x;
    const int row0 = blockIdx.x * ROWS_PER_BLOCK;
    const uint32_t ldsBase = (uint32_t)(uintptr_t)&s_buf[0];   // low 32 bits = LDS byte offset

    float acc[CHUNKS * 4];
#pragma unroll
    for (int k = 0; k < CHUNKS * 4; ++k) acc[k] = 0.0f;

    // Prologue: stage first row.
    async_row_load(ts, row0, ldsBase, tid);

    for (int r = 0; r < ROWS_PER_BLOCK; ++r) {
        const int row = row0 + r;

        // Wait for this row's async copies (in-order completion; private regions,
        // so no workgroup barrier needed for staging).
        asm volatile("s_wait_asynccnt 0x0" ::: "memory");

        if (tid == 0) {
            float dv  = ts[(size_t)row * NN + row];
            s_misc[0] = dv * dv;               // diag[row]
            s_misc[1] = state[3 * row + 2];    // V[row]
        }

        // Read my 32 columns from LDS, square, partial row sum.
        float x[CHUNKS * 4];
        float psum = 0.0f;
#pragma unroll
        for (int j = 0; j < CHUNKS; ++j) {
            float4 v4 = *reinterpret_cast<const float4*>(&s_buf[j * 1024 + tid * 4]);
            x[4 * j + 0] = v4.x; x[4 * j + 1] = v4.y;
            x[4 * j + 2] = v4.z; x[4 * j + 3] = v4.w;
            psum = fmaf(v4.x, v4.x, psum);
            psum = fmaf(v4.y, v4.y, psum);
            psum = fmaf(v4.z, v4.z, psum);
            psum = fmaf(v4.w, v4.w, psum);
        }

        // Buffer is now consumed (values live in x[]): prefetch next row into the
        // same buffer. Ensure DS reads fully retired before async writes can land.
        if (r + 1 < ROWS_PER_BLOCK) {
            asm volatile("s_wait_dscnt 0x0" ::: "memory");
            async_row_load(ts, row + 1, ldsBase, tid);
        }

        s_red[tid] = psum;
        __syncthreads();

        // Every wave (fully active -> EXEC all-1s, as WMMA requires) folds 256 -> 32,
        // then reduces 32 -> scalar on the matrix pipe.
        const int lane = tid & 31;
        float v = 0.0f;
#pragma unroll
        for (int k = 0; k < 8; ++k) v += s_red[lane + 32 * k];

        float row_sum;
#if __has_builtin(__builtin_amdgcn_wmma_f32_16x16x4_f32)
        {
            // A(16x4): VGPR0 = v (A[m][0] lanes 0-15, A[m][2] lanes 16-31), VGPR1 = 0.
            // B = ones -> D[m][n] = v_m + v_{m+16}, replicated over n.
            v2f a = {v, 0.0f};
            v2f b = {1.0f, 1.0f};
            v8f c = {};
            v8f d = __builtin_amdgcn_wmma_f32_16x16x4_f32(
                /*neg_a=*/false, a, /*neg_b=*/false, b,
                /*c_mod=*/(short)0, c, /*reuse_a=*/false, /*reuse_b=*/false);
            float s = d[0] + d[1] + d[2] + d[3] + d[4] + d[5] + d[6] + d[7];
            row_sum = s + __shfl_xor(s, 16, 32);   // halves hold rows 0-7 / 8-15
        }
#else
        {
            float s = v;
#pragma unroll
            for (int off = 16; off >= 1; off >>= 1) s += __shfl_xor(s, off, 32);
            row_sum = s;
        }
#endif

        const float dsq  = s_misc[0];
        const float Vr   = s_misc[1];
        const float nrm  = fminf(dsq / (row_sum - dsq), 1.0f);
        const float w    = nrm * Vr;
        if (tid == 0) {
            g_norm[row] = nrm;
            g_diag[row] = dsq;
        }

        // coupling partials stay in registers: acc[j] += w * x^2
#pragma unroll
        for (int k = 0; k < CHUNKS * 4; ++k) acc[k] = fmaf(w * x[k], x[k], acc[k]);

        __syncthreads();   // protect s_red / s_misc reuse next iteration
    }

    // Flush per-block coupling partials (512 adders per address -> cheap L2 atomics).
#pragma unroll
    for (int j = 0; j < CHUNKS; ++j) {
#pragma unroll
        for (int e = 0; e < 4; ++e) {
            atomicAdd(&g_coupling[j * 1024 + tid * 4 + e], acc[4 * j + e]);
        }
    }
}

__global__ void zero_coupling_kernel(float* __restrict__ g_coupling) {
    int i = blockIdx.x * blockDim.x + threadIdx.x;
    if (i < NN) g_coupling[i] = 0.0f;
}

__global__ void combine_kernel(const float* __restrict__ state,
                               const float* __restrict__ betas,
                               const float* __restrict__ deltas,
                               const float* __restrict__ cs,
                               const float* __restrict__ ps,
                               const float* __restrict__ g_coupling,
                               const float* __restrict__ g_norm,
                               const float* __restrict__ g_diag,
                               float* __restrict__ out) {
    int j = blockIdx.x * blockDim.x + threadIdx.x;
    if (j >= NN) return;
    float U = state[3 * j + 0];
    float I = state[3 * j + 1];
    float V = state[3 * j + 2];
    float b2 = betas[j] * betas[j];
    float d2 = deltas[j] * deltas[j];
    float p2 = ps[j] * ps[j];
    float c2 = cs[j] * cs[j];
    float bUV = b2 * U * V;
    float diag = g_diag[j];
    float nrm  = g_norm[j];
    float coupling = g_coupling[j] - nrm * diag * V;
    float dV = p2 * I - c2 * V - diag + coupling;
    out[3 * j + 0] = -bUV;
    out[3 * j + 1] = bUV - d2 * I;
    out[3 * j + 2] = dV;
}

extern "C" void kernel_launch(void* const* d_in, const int* in_sizes, int n_in,
                              void* d_out, int out_size, void* d_ws, size_t ws_size,
                              hipStream_t stream) {
    // inputs: 0=t (unused), 1=state(3N), 2=betas, 3=deltas, 4=cs, 5=ps, 6=ts(NxN)
    const float* state  = (const float*)d_in[1];
    const float* betas  = (const float*)d_in[2];
    const float* deltas = (const float*)d_in[3];
    const float* cs     = (const float*)d_in[4];
    const float* ps     = (const float*)d_in[5];
    const float* ts     = (const float*)d_in[6];
    float* out = (float*)d_out;

    float* g_coupling = (float*)d_ws;          // [0, N)
    float* g_norm     = g_coupling + NN;       // [N, 2N)
    float* g_diag     = g_norm + NN;           // [2N, 3N)

    zero_coupling_kernel<<<NN / 256, 256, 0, stream>>>(g_coupling);
    fused_rows_kernel<<<NBLOCKS, THREADS, 0, stream>>>(state, ts, g_coupling, g_norm, g_diag);
    combine_kernel<<<NN / 256, 256, 0, stream>>>(state, betas, deltas, cs, ps,
                                                 g_coupling, g_norm, g_diag, out);
}